// HeteroTripartiteGCN_37941741093201
// MI455X (gfx1250) — compile-verified
//
#include <hip/hip_runtime.h>

#define NU 100000
#define NV 50000
#define NF 5000
#define DD 64

typedef __attribute__((ext_vector_type(2))) float v2f;
typedef __attribute__((ext_vector_type(8))) float v8f;
typedef __attribute__((ext_vector_type(4))) unsigned int u32x4;
typedef __attribute__((ext_vector_type(8))) int i32x8;
typedef __attribute__((ext_vector_type(4))) int i32x4;

#if defined(__HIP_DEVICE_COMPILE__) && __has_builtin(__builtin_amdgcn_tensor_load_to_lds) && \
    __has_builtin(__builtin_amdgcn_s_wait_tensorcnt)
#define USE_TDM 1
#endif

// Guaranteed hardware fp32 atomic add (L2 RMW), no CAS loop.
__device__ __forceinline__ void atomic_add_f32(float* p, float v) {
    asm volatile("global_atomic_add_f32 %0, %1, off" :: "v"(p), "v"(v) : "memory");
}

#ifdef USE_TDM
typedef __attribute__((address_space(3))) void las_void;

// 32-bit LDS byte offset of a __shared__ object (generic -> AS3 -> int folds to
// the literal LDS offset).
__device__ __forceinline__ unsigned lds_off(void* p) {
    return (unsigned)(unsigned long long)(las_void*)p;
}

// Issue one TDM 2D tile load: global (row-major, stride0 elements/row) -> LDS.
// Rows past tensor_d1 are OOB and zero-filled by the TDM. Optional LDS padding
// of pad_dwords after every interval_dwords (power of two, >= 2).
__device__ __forceinline__ void tdm_load_2d(unsigned lds_addr, const void* gaddr,
                                            unsigned tensor_d0, unsigned tensor_d1,
                                            unsigned tile_d0, unsigned tile_d1,
                                            unsigned stride0,
                                            int pad_interval_code, int pad_amount_code) {
    const unsigned long long ga = (unsigned long long)gaddr;
    u32x4 g0;
    g0.x = 0x1u;                                            // count=1, user descriptor
    g0.y = lds_addr;                                        // lds_addr [63:32]
    g0.z = (unsigned)(ga & 0xFFFFFFFFu);                    // global_addr lo
    g0.w = (unsigned)((ga >> 32) & 0x1FFFFFFu) | (2u << 30);// global_addr hi | type=2
    i32x8 g1;
    unsigned w0 = (2u << 16);                               // data_size = 4 bytes
    if (pad_interval_code >= 0)
        w0 |= (1u << 20) | ((unsigned)pad_interval_code << 22) |
              ((unsigned)pad_amount_code << 25);            // pad_enable/interval/amount
    g1[0] = (int)w0;
    g1[1] = (int)((tensor_d0 & 0xFFFFu) << 16);             // dim0[15:0] in [63:48]
    g1[2] = (int)(((tensor_d0 >> 16) & 0xFFFFu) | ((tensor_d1 & 0xFFFFu) << 16));
    g1[3] = (int)(((tensor_d1 >> 16) & 0xFFFFu) | ((tile_d0 & 0xFFFFu) << 16));
    g1[4] = (int)(tile_d1 & 0xFFFFu);                       // tile_dim1; tile_dim2=0
    g1[5] = (int)stride0;                                   // tensor_dim0_stride lo32
    g1[6] = 0;                                              // stride0 hi16 | stride1 lo16
    g1[7] = 0;                                              // stride1 hi32
    const i32x4 z4 = {0, 0, 0, 0};                          // groups 2/3 unused (2D)
    const i32x8 z8 = {0, 0, 0, 0, 0, 0, 0, 0};              // unused trailing group
    __builtin_amdgcn_tensor_load_to_lds(g0, g1, z4, z4, z8, 0);
}
#endif

// ---------------------------------------------------------------------------
// Y = X @ W, X: nrows x 64, W: 64 x 64 (row-major), Y: nrows x 64. fp32 WMMA.
// 256 threads = 8 wave32; each wave computes a 16x64 output tile.
// Tiles are staged into LDS by the Tensor Data Mover (wave 0 issues, EXEC is
// ignored by TDM so the gate must be a *scalar* branch via readfirstlane).
// ---------------------------------------------------------------------------
#define XS 68  // LDS row stride (floats): 64 + 4 pad dwords -> spreads banks

__global__ __launch_bounds__(256) void gemm64_wmma(const float* __restrict__ X,
                                                   const float* __restrict__ W,
                                                   float* __restrict__ Y,
                                                   int nrows) {
    __shared__ float sW[64 * 64];
    __shared__ float sX[128 * XS];

    const int t    = threadIdx.x;
    const int row0 = blockIdx.x * 128;

#ifdef USE_TDM
    // Scalar-uniform gate: only wave 0 issues the two DMA descriptors.
    if (__builtin_amdgcn_readfirstlane(t >> 5) == 0) {
        // W: 64x64 tile, contiguous, no LDS padding.
        tdm_load_2d(lds_off(sW), W, 64, 64, 64, 64, 64, -1, 0);
        // X: 128x64 tile; rows past (nrows - row0) zero-filled by TDM OOB;
        // LDS pad 4 dwords every 64 dwords -> row stride XS=68 floats.
        tdm_load_2d(lds_off(sX), X + (size_t)row0 * DD,
                    64, (unsigned)(nrows - row0), 64, 128, 64,
                    /*interval 64 dwords=*/5, /*amount 4 dwords=*/3);
        __builtin_amdgcn_s_wait_tensorcnt(0);
    }
#else
    for (int i = t * 4; i < 64 * 64; i += 256 * 4)
        *(float4*)(sW + i) = *(const float4*)(W + i);
    for (int i = t * 4; i < 128 * 64; i += 256 * 4) {
        const int r = i >> 6, c = i & 63;
        float4 val = make_float4(0.f, 0.f, 0.f, 0.f);
        if (row0 + r < nrows) val = *(const float4*)(X + (size_t)(row0 + r) * DD + c);
        *(float4*)(sX + r * XS + c) = val;
    }
#endif
    __syncthreads();

    const int wave = t >> 5;        // 0..7 -> rows row0 + 16*wave ..
    const int lane = t & 31;
    const int l16  = lane & 15;
    const int hi   = lane >> 4;     // lanes 16-31 hold the upper K pair / M+8 rows
    const float* aRow = sX + (wave * 16 + l16) * XS;

    #pragma unroll
    for (int n = 0; n < 4; ++n) {   // four 16-col output tiles
        v8f c = {0.f, 0.f, 0.f, 0.f, 0.f, 0.f, 0.f, 0.f};
        #pragma unroll
        for (int kk = 0; kk < 16; ++kk) {   // K = 64 in steps of 4
            const int k0 = kk * 4 + hi * 2;
            // A 16x4 f32 layout: lane l<16 -> A[l][k0..k0+1]; lanes 16-31 -> K+2
            v2f a = *(const v2f*)(aRow + k0);
            // B 4x16 f32 layout mirrors C/D striping: vgpr j -> K=j (lo) / K=j+2 (hi)
            v2f b;
            b.x = sW[(k0 + 0) * 64 + n * 16 + l16];
            b.y = sW[(k0 + 1) * 64 + n * 16 + l16];
            c = __builtin_amdgcn_wmma_f32_16x16x4_f32(
                    /*neg_a=*/false, a, /*neg_b=*/false, b,
                    /*c_mod=*/(short)0, c, /*reuse_a=*/false, /*reuse_b=*/false);
        }
        // C/D layout: vgpr j, lanes 0-15 -> M=j, lanes 16-31 -> M=j+8; N = lane%16
        #pragma unroll
        for (int j = 0; j < 8; ++j) {
            const int grow = row0 + wave * 16 + j + hi * 8;
            if (grow < nrows) Y[(size_t)grow * DD + n * 16 + l16] = c[j];
        }
    }
}

// ---------------------------------------------------------------------------
// Edge-parallel COO SpMM scatter: out[r,:] += v * x[c,:].  16 lanes per edge,
// each lane moves 4 consecutive floats (float4 gather + 4 HW fp32 atomics).
// Gathered matrices and outputs are L2-resident (192 MB), so atomics resolve
// in L2 without HBM round-trips.
// ---------------------------------------------------------------------------
__global__ __launch_bounds__(256) void spmm_edge_atomic(const int* __restrict__ rows,
                                                        const int* __restrict__ cols,
                                                        const float* __restrict__ vals,
                                                        const float* __restrict__ x,
                                                        float* __restrict__ out,
                                                        int nE) {
    const long long t = (long long)blockIdx.x * 256 + threadIdx.x;
    const int e = (int)(t >> 4);
    if (e >= nE) return;
    const int q = ((int)t & 15) * 4;

    const int   r = rows[e];
    const int   c = cols[e];
    const float v = vals[e];

    const float4 xv = *(const float4*)(x + (size_t)c * DD + q);
    float* o = out + (size_t)r * DD + q;
    atomic_add_f32(o + 0, v * xv.x);
    atomic_add_f32(o + 1, v * xv.y);
    atomic_add_f32(o + 2, v * xv.z);
    atomic_add_f32(o + 3, v * xv.w);
}

__global__ __launch_bounds__(256) void relu_inplace(float* __restrict__ p, int n) {
    const int i = blockIdx.x * 256 + threadIdx.x;
    if (i < n) {
        const float v = p[i];
        p[i] = v > 0.f ? v : 0.f;
    }
}

// ---------------------------------------------------------------------------
extern "C" void kernel_launch(void* const* d_in, const int* in_sizes, int n_in,
                              void* d_out, int out_size, void* d_ws, size_t ws_size,
                              hipStream_t stream) {
    const float* x_u    = (const float*)d_in[0];
    const float* x_v    = (const float*)d_in[1];
    const float* x_f    = (const float*)d_in[2];
    const float* W_u_uv = (const float*)d_in[3];
    const float* W_v_uv = (const float*)d_in[4];
    const float* W_f2u  = (const float*)d_in[5];
    const float* W_f2v  = (const float*)d_in[6];
    const float* W_u2f  = (const float*)d_in[7];
    const float* W_v2f  = (const float*)d_in[8];

    // Workspace layout (floats): six transformed feature matrices (~79 MB)
    float* ws     = (float*)d_ws;
    float* tmp_u  = ws;                              // NU*64
    float* tmp_v  = tmp_u + (size_t)NU * DD;         // NV*64
    float* u2f    = tmp_v + (size_t)NV * DD;         // NU*64
    float* v2f_m  = u2f   + (size_t)NU * DD;         // NV*64
    float* f2u    = v2f_m + (size_t)NV * DD;         // NF*64
    float* f2v    = f2u   + (size_t)NF * DD;         // NF*64

    // Zero the output accumulators (graph-capturable memset)
    (void)hipMemsetAsync(d_out, 0, (size_t)out_size * sizeof(float), stream);

    const dim3 blk(256);
    // Dense fp32 WMMA transforms (TDM-staged tiles)
    gemm64_wmma<<<(NU + 127) / 128, blk, 0, stream>>>(x_u, W_u_uv, tmp_u, NU);
    gemm64_wmma<<<(NV + 127) / 128, blk, 0, stream>>>(x_v, W_v_uv, tmp_v, NV);
    gemm64_wmma<<<(NU + 127) / 128, blk, 0, stream>>>(x_u, W_u2f,  u2f,   NU);
    gemm64_wmma<<<(NV + 127) / 128, blk, 0, stream>>>(x_v, W_v2f,  v2f_m, NV);
    gemm64_wmma<<<(NF + 127) / 128, blk, 0, stream>>>(x_f, W_f2u,  f2u,   NF);
    gemm64_wmma<<<(NF + 127) / 128, blk, 0, stream>>>(x_f, W_f2v,  f2v,   NF);

    float* msg_u = (float*)d_out;
    float* msg_v = msg_u + (size_t)NU * DD;
    float* msg_f = msg_v + (size_t)NV * DD;

    // SpMM scatter phases (edge counts from harness-provided sizes)
    struct Sp { int ri, ci, vi; const float* x; float* o; };
    const Sp sp[6] = {
        { 9, 10, 11, tmp_v, msg_u},   // uv:  msg_u += A_uv @ tmp_v
        {12, 13, 14, tmp_u, msg_v},   // vu:  msg_v += A_vu @ tmp_u
        {15, 16, 17, f2u,   msg_u},   // uf:  msg_u += A_uf @ f2u
        {18, 19, 20, f2v,   msg_v},   // vf:  msg_v += A_vf @ f2v
        {21, 22, 23, u2f,   msg_f},   // fu:  msg_f += A_fu @ u2f
        {24, 25, 26, v2f_m, msg_f},   // fv:  msg_f += A_fv @ v2f
    };
    for (int s = 0; s < 6; ++s) {
        const int nE = in_sizes[sp[s].vi];
        const long long threads = (long long)nE * 16;
        const int grid = (int)((threads + 255) / 256);
        spmm_edge_atomic<<<grid, blk, 0, stream>>>(
            (const int*)d_in[sp[s].ri], (const int*)d_in[sp[s].ci],
            (const float*)d_in[sp[s].vi], sp[s].x, sp[s].o, nE);
    }

    relu_inplace<<<(out_size + 255) / 256, blk, 0, stream>>>((float*)d_out, out_size);
}